// GraphConvolution_60902636257281
// MI455X (gfx1250) — compile-verified
//
#include <hip/hip_runtime.h>

#define D 128            // feature dim (both in and out)
#define APAD 4           // LDS row padding (floats): keeps 8B alignment, spreads banks
#define ASTRIDE (D + APAD)

typedef __attribute__((ext_vector_type(2))) float v2f;
typedef __attribute__((ext_vector_type(8))) float v8f;

// ---------------------------------------------------------------------------
// Kernel 1: hi = SpMM(adj, x);  support = (1-alpha)*hi + alpha*h0
// One 128-thread block per node. adj_rows is sorted, so thread 0 binary-
// searches the edge range [start,end) for this node and broadcasts via LDS.
// Per-edge cols[e]/vals[e] are wave-uniform -> scalar loads; x-gather is
// coalesced across the 128 feature lanes (L2-resident working set).
// ---------------------------------------------------------------------------
__global__ __launch_bounds__(128) void spmm_support_kernel(
    const float* __restrict__ x, const float* __restrict__ h0,
    const int* __restrict__ rows, const int* __restrict__ cols,
    const float* __restrict__ vals, const float* __restrict__ alpha_p,
    float* __restrict__ support, int nE) {
  __shared__ int se[2];
  const int node = blockIdx.x;
  if (threadIdx.x == 0) {
    // lower_bound(rows, node)
    int lo = 0, hi = nE;
    while (lo < hi) {
      int mid = (lo + hi) >> 1;
      if (rows[mid] < node) lo = mid + 1; else hi = mid;
    }
    se[0] = lo;
    // upper_bound(rows, node) -- continue from lo
    hi = nE;
    while (lo < hi) {
      int mid = (lo + hi) >> 1;
      if (rows[mid] <= node) lo = mid + 1; else hi = mid;
    }
    se[1] = lo;
  }
  __syncthreads();
  const int start = se[0];
  const int end   = se[1];
  const int f = threadIdx.x;

  float acc = 0.0f;
  for (int e = start; e < end; ++e) {
    acc = fmaf(vals[e], x[(size_t)cols[e] * D + f], acc);
  }
  const float alpha = alpha_p[0];
  const size_t idx = (size_t)node * D + f;
  support[idx] = (1.0f - alpha) * acc + alpha * h0[idx];
}

// ---------------------------------------------------------------------------
// Kernel 2: out = theta*(support @ W) + (1-theta)*support + x
// One block (8 waves) per 16-row strip of support. Strip staged in LDS
// (padded stride -> conflict-free ds_load_b64); wave w computes N-tile w with
// 32x V_WMMA_F32_16X16X4_F32 (fp32 end-to-end, matches reference precision).
//
// VGPR layouts per CDNA5 ISA 7.12.2:
//   A 16x4 f32:  lane m = lane&15 holds row m; lane[4] selects k-pair {0,1}/{2,3}
//   B 4x16 f32:  lane n = lane&15 holds col n; lane[4] selects k-pair
//   C/D 16x16:   vgpr i: lanes 0-15 -> M=i, lanes 16-31 -> M=i+8; N = lane&15
// ---------------------------------------------------------------------------
__global__ __launch_bounds__(256) void gemm_out_kernel(
    const float* __restrict__ support, const float* __restrict__ weight,
    const float* __restrict__ x, const float* __restrict__ lamda_p,
    const int* __restrict__ layer_p, float* __restrict__ out, int nN) {
  __shared__ float alds[16 * ASTRIDE];   // 16 x 128 strip of support (8.25 KB)

  const int mtile = blockIdx.x;          // 16-row strip index
  const int mbase = mtile * 16;

  // Cooperative stage of A strip (coalesced 128B-per-row loads).
  for (int t = threadIdx.x; t < 16 * D; t += 256) {
    const int r = t >> 7;                // 0..15
    const int c = t & (D - 1);           // 0..127
    alds[r * ASTRIDE + c] = support[(size_t)(mbase + r) * D + c];
  }
  // Prefetch next strip into the cache hierarchy while we compute.
  if (mtile + 1 < nN / 16) {
    __builtin_prefetch(support + (size_t)(mbase + 16) * D + (threadIdx.x & 31) * 16, 0, 0);
  }
  __syncthreads();

  const int wave  = threadIdx.x >> 5;    // 0..7 -> N tile
  const int lane  = threadIdx.x & 31;
  const int mn    = lane & 15;           // row (A) / col (B) within tile
  const int kh    = (lane >> 4) << 1;    // k-pair base: 0 or 2
  const int ncol  = wave * 16 + mn;      // absolute output column 0..127

  v8f c = {};
#pragma unroll
  for (int k0 = 0; k0 < D; k0 += 4) {
    // A: rows of the LDS strip, two consecutive k values (8B-aligned ds load)
    v2f a = *(const v2f*)(&alds[mn * ASTRIDE + k0 + kh]);
    // B: weight is 64KB -> WGP$/L2 resident; lanes 0-15 / 16-31 each coalesce
    v2f b;
    b.x = weight[(size_t)(k0 + kh)     * D + ncol];
    b.y = weight[(size_t)(k0 + kh + 1) * D + ncol];
    c = __builtin_amdgcn_wmma_f32_16x16x4_f32(
        /*neg_a=*/false, a, /*neg_b=*/false, b,
        /*c_mod=*/(short)0, c, /*reuse_a=*/false, /*reuse_b=*/false);
  }

  // theta = log(lamda/layer_idx + 1)  (uniform scalar loads)
  const float theta = __logf(lamda_p[0] / (float)layer_p[0] + 1.0f);
  const float omt   = 1.0f - theta;

  // Epilogue: out = theta*acc + (1-theta)*support + x
  const int rsel = (lane >> 4) << 3;     // 0 or 8
#pragma unroll
  for (int i = 0; i < 8; ++i) {
    const int rloc = rsel + i;           // 0..15 within strip
    const size_t idx = (size_t)(mbase + rloc) * D + ncol;
    out[idx] = theta * c[i] + omt * alds[rloc * ASTRIDE + ncol] + x[idx];
  }
}

extern "C" void kernel_launch(void* const* d_in, const int* in_sizes, int n_in,
                              void* d_out, int out_size, void* d_ws, size_t ws_size,
                              hipStream_t stream) {
  const float* x      = (const float*)d_in[0];
  const float* h0     = (const float*)d_in[1];
  const float* weight = (const float*)d_in[2];
  const float* lamda  = (const float*)d_in[3];
  const float* alpha  = (const float*)d_in[4];
  const int*   rows   = (const int*)d_in[5];
  const int*   cols   = (const int*)d_in[6];
  const float* vals   = (const float*)d_in[7];
  const int*   layer  = (const int*)d_in[8];
  float* out = (float*)d_out;

  const int nN = in_sizes[0] / D;        // 50000
  const int nE = in_sizes[5];            // 800000

  float* support = (float*)d_ws;         // nN * D floats (25.6 MB) in workspace

  spmm_support_kernel<<<nN, 128, 0, stream>>>(x, h0, rows, cols, vals, alpha,
                                              support, nE);
  gemm_out_kernel<<<nN / 16, 256, 0, stream>>>(support, weight, x, lamda,
                                               layer, out, nN);
}